// Convolution_730144440357
// MI455X (gfx1250) — compile-verified
//
#include <hip/hip_runtime.h>

// Sparse 3x3x3 conv, fp32, via V_WMMA_F32_16X16X4_F32 on gfx1250 (wave32).
//
// out[q, f] = sum_{k in 27 taps} sum_c W[k][c][f] * feat[n(q,k)][c]
// Scatter form: for each point n, tap d: target cell lid_in[n] - lin(d);
// locate row in sorted lid_out; atomic-add the W[k]-projected features.
// lid_out is strictly increasing, so the 3 dz-taps of a (dx,dy) group have
// consecutive targets -> 1 binary search + <=2-step scan-down per group.

#define GRIDC 256
#define CIN   32
#define FOUT  32
#define NTAP  27

typedef __attribute__((ext_vector_type(2))) float v2f;
typedef __attribute__((ext_vector_type(8))) float v8f;

// Types for the async-to-LDS builtin (diagnostic showed GCC vector_size(16)
// int pointee in device/global AS for arg0).
typedef int v4i_gcc __attribute__((vector_size(16)));
typedef __attribute__((address_space(1))) v4i_gcc* gv4i_p;   // global AS
typedef __attribute__((address_space(3))) v4i_gcc* lv4i_p;   // LDS AS

#define SWZ_F2   (NTAP * 8 * 2 * 32)          // 13824 v2f : swizzled B operands
#define RAW_F    (NTAP * CIN * FOUT)          // 27648 f32 : raw W staging
#define LDS_BYTES (SWZ_F2 * 8 + RAW_F * 4)    // 110592 + 110592 = 221184 B

// ---------------------------------------------------------------- helpers ---
__global__ void __launch_bounds__(256) zero_f32(float* __restrict__ p, int n) {
    int i = blockIdx.x * blockDim.x + threadIdx.x;
    if (i < n) p[i] = 0.0f;
}

__global__ void __launch_bounds__(256) lin_out_pos(const int* __restrict__ op,
                                                   int* __restrict__ lid, int M) {
    int i = blockIdx.x * blockDim.x + threadIdx.x;
    if (i < M) {
        int x = op[3 * i + 0], y = op[3 * i + 1], z = op[3 * i + 2];
        lid[i] = (x * GRIDC + y) * GRIDC + z;
    }
}

// ------------------------------------------------------------ main kernel ---
// 256 threads = 8 waves. Wave w of block b handles the 16-point input tile
// (b*8 + w).
//
// LDS region 0 (wlds, 13824 v2f): W pre-swizzled into the WMMA B-operand
// layout. Per (k, kk, h): 32 lanes x float2, lane L (L<16) holds
// (W[k][4kk+0][16h+L], W[k][4kk+1][16h+L]); lane L+16 holds
// (W[k][4kk+2][16h+L], W[k][4kk+3][16h+L]).  Index: ((k*8+kk)*2+h)*32 + lane.
// LDS region 1 (rawW): byte-copy of W, filled by async-to-LDS DMA when
// available, used only as the swizzle source.
__global__ void __launch_bounds__(256) sparse_conv_wmma(
    const float* __restrict__ feat,     // [N, 32]
    const int*   __restrict__ inpos,    // [N, 3]
    const float* __restrict__ W,        // [27, 32, 32]
    const int*   __restrict__ lid_out,  // [M] strictly increasing
    float*       __restrict__ out,      // [M, 32] (pre-zeroed)
    int N, int M)
{
    extern __shared__ v2f wlds[];               // see LDS_BYTES
    float* rawW = (float*)(wlds + SWZ_F2);

    const int tid     = threadIdx.x;
    const int lane    = tid & 31;
    const int wave    = tid >> 5;
    const int lane15  = lane & 15;
    const int halfSel = lane >> 4;              // 0: K=0,1 half  1: K=2,3 half

    // ---- stage raw W into LDS ------------------------------------------
#if __has_builtin(__builtin_amdgcn_global_load_async_to_lds_b128)
    for (int ofs = tid * 16; ofs < RAW_F * 4; ofs += 256 * 16) {
        __builtin_amdgcn_global_load_async_to_lds_b128(
            (gv4i_p)((const char*)W + ofs),
            (lv4i_p)((char*)rawW + ofs),
            0, 0);
    }
#if __has_builtin(__builtin_amdgcn_s_wait_asynccnt)
    __builtin_amdgcn_s_wait_asynccnt(0);
#else
    asm volatile("s_wait_asynccnt 0x0" ::: "memory");
#endif
#else
    for (int t = tid; t < RAW_F; t += 256) rawW[t] = W[t];
#endif
    __syncthreads();

    // ---- swizzle raw W -> B-operand layout (LDS -> LDS) ----------------
    for (int t = tid; t < SWZ_F2; t += 256) {
        int l  = t & 31;
        int h  = (t >> 5) & 1;
        int kk = (t >> 6) & 7;
        int k  = t >> 9;
        int c  = 4 * kk + 2 * (l >> 4);
        int f  = h * 16 + (l & 15);
        v2f v;
        v.x = rawW[(k * CIN + c) * FOUT + f];
        v.y = rawW[(k * CIN + c + 1) * FOUT + f];
        wlds[t] = v;
    }
    __syncthreads();

    const int tile = blockIdx.x * 8 + wave;
    const int p0   = tile * 16;
    if (p0 >= N) return;

    const int  prow    = p0 + lane15;
    const bool inrange = (prow < N);
    const int  prowc   = inrange ? prow : (N - 1);

    // ---- per-point position (lanes 16-31 mirror lanes 0-15) ------------
    const int px = inpos[prowc * 3 + 0];
    const int py = inpos[prowc * 3 + 1];
    const int pz = inpos[prowc * 3 + 2];
    const int lid_in = (px * GRIDC + py) * GRIDC + pz;

    // ---- A-matrix: 8 K-slabs of the 16x32 feature tile -----------------
    // ISA 16x4 fp32 A layout: lane L<16 -> (K=0,K=1), lane L+16 -> (K=2,K=3)
    v2f a[8];
#pragma unroll
    for (int kk = 0; kk < 8; ++kk) {
        const float* src = feat + (size_t)prowc * CIN + 4 * kk + 2 * halfSel;
        a[kk] = *(const v2f*)src;               // global_load_b64, 8B-aligned
    }

    // ---- 9 (dx,dy) groups; 1 search + scan-down covers dz = 0,1,2 ------
    for (int g = 0; g < 9; ++g) {
        const int dx = g / 3, dy = g % 3;
        const int t0 = lid_in - (dx * GRIDC + dy) * GRIDC;   // dz = 0 target

        // lower_bound(lid_out, t0)
        int lo = 0, hi = M;
        while (lo < hi) {
            int mid = (lo + hi) >> 1;
            if (lid_out[mid] < t0) lo = mid + 1; else hi = mid;
        }
        int j = lo;

        for (int dz = 0; dz < 3; ++dz) {
            const int k = g * 3 + dz;
            const int t = t0 - dz;
            // strictly-increasing lid_out => lower_bound(t0-dz) >= j-dz
            while (j > 0 && lid_out[j - 1] >= t) --j;

            const bool inb   = inrange && (px >= dx) && (py >= dy) && (pz >= dz);
            const int  idx   = (j < M) ? j : (M - 1);
            const bool valid = inb && (j < M) && (lid_out[idx] == t);

            // ---- D[16x32] = A[16x32] @ W_k[32x32], two 16-wide N halves
            v8f acc0 = {}; v8f acc1 = {};
            const v2f* wk = wlds + k * (8 * 2 * 32);
#pragma unroll
            for (int kk = 0; kk < 8; ++kk) {
                v2f b0 = wk[kk * 64 + 0 * 32 + lane];   // ds_load_b64, N half 0
                v2f b1 = wk[kk * 64 + 1 * 32 + lane];   // ds_load_b64, N half 1
                acc0 = __builtin_amdgcn_wmma_f32_16x16x4_f32(
                    false, a[kk], false, b0, (short)0, acc0, false, false);
                acc1 = __builtin_amdgcn_wmma_f32_16x16x4_f32(
                    false, a[kk], false, b1, (short)0, acc1, false, false);
            }

            // ---- scatter: lane L<16 rows M=0..7 (col L), L+16 rows 8..15
#pragma unroll
            for (int r = 0; r < 8; ++r) {
                const int m      = r + 8 * halfSel;       // tile-local point
                const int ridx   = __shfl(idx, m, 32);    // from lane m (0..15)
                const int rvalid = __shfl((int)valid, m, 32);
                if (rvalid) {
                    float* o = out + (size_t)ridx * FOUT;
                    atomicAdd(o + lane15,      acc0[r]);  // global_atomic_add_f32
                    atomicAdd(o + 16 + lane15, acc1[r]);
                }
            }
        }
    }
}

// --------------------------------------------------------------- launch ----
extern "C" void kernel_launch(void* const* d_in, const int* in_sizes, int n_in,
                              void* d_out, int out_size, void* d_ws, size_t ws_size,
                              hipStream_t stream) {
    const float* feat   = (const float*)d_in[0];   // [N,32] fp32
    const int*   inpos  = (const int*)  d_in[1];   // [N,3]  int32
    const int*   outpos = (const int*)  d_in[2];   // [M,3]  int32
    const float* W      = (const float*)d_in[3];   // [27,32,32] fp32

    const int N = in_sizes[0] / CIN;
    const int M = in_sizes[2] / 3;

    int*   lid = (int*)d_ws;        // [M] sorted linearized out cells
    float* out = (float*)d_out;     // [M,32]

    // zero output (harness poisons it)
    zero_f32<<<(out_size + 255) / 256, 256, 0, stream>>>(out, out_size);

    // linearize out_positions (already lexicographically sorted by np.unique)
    lin_out_pos<<<(M + 255) / 256, 256, 0, stream>>>(outpos, lid, M);

    // main conv: 1 wave per 16-point tile, 8 waves per block
    const int tiles  = (N + 15) / 16;
    const int blocks = (tiles + 7) / 8;
    hipLaunchKernelGGL(sparse_conv_wmma, dim3(blocks), dim3(256),
                       (size_t)LDS_BYTES, stream,
                       feat, inpos, W, lid, out, N, M);
}